// MultiCNNAttention_26568667693701
// MI455X (gfx1250) — compile-verified
//
#include <hip/hip_runtime.h>

typedef __attribute__((ext_vector_type(2))) float v2f;
typedef __attribute__((ext_vector_type(8))) float v8f;

#define LDS_FENCE()   asm volatile("s_wait_dscnt 0x0" ::: "memory")
#define ASYNC_FENCE() asm volatile("s_wait_asynccnt 0x0" ::: "memory")

// Async 16B global->LDS DMA (CDNA5, tracked by ASYNCcnt).
__device__ __forceinline__ void async_copy16(unsigned ldsa, unsigned long long ga) {
  asm volatile("global_load_async_to_lds_b128 %0, %1, off" :: "v"(ldsa), "v"(ga) : "memory");
}
__device__ __forceinline__ unsigned lds_off(const void* p) {
  return (unsigned)(uintptr_t)p;
}
__device__ __forceinline__ unsigned long long gaddr(const void* p) {
  return (unsigned long long)(uintptr_t)p;
}

// Problem constants (reference: B=2, S=2048, E=1024, H=16, D=64)
constexpr int cB = 2, cS = 2048, cE = 1024, cH = 16, cD = 64;

// ---------------------------------------------------------------------------
// GEMM with bias: C[M,N] = A[M,K] @ W[K,N] + bias[N]
// 256 threads = 8 waves. Block tile 64x128, K tile 32, double-buffered LDS:
// async DMA for K-block k+1 runs under the WMMA chain of K-block k.
// Wave grid 4(M) x 2(N): each wave 16 rows x 64 cols = 4 f32 WMMA accums.
// ---------------------------------------------------------------------------
__global__ __launch_bounds__(256) void gemm_bias_kernel(
    const float* __restrict__ A, const float* __restrict__ W,
    const float* __restrict__ bias, float* __restrict__ C,
    int M, int N, int K)
{
  __shared__ float As[2][64 * 36];    // stride 36: conflict-free column reads
  __shared__ float Bs[2][32 * 132];   // stride 132: conflict-free row reads

  const int tid  = threadIdx.x;
  const int lane = tid & 31;
  const int ln   = lane & 15;
  const int lh   = lane >> 4;
  const int wv   = tid >> 5;
  const int wr   = wv >> 1;           // 0..3
  const int wc   = wv & 1;            // 0..1
  const int bm   = blockIdx.y * 64;
  const int bn   = blockIdx.x * 128;

  v8f acc[4] = {};

  auto stage = [&](int buf, int k0) {
#pragma unroll
    for (int i = 0; i < 2; ++i) {     // A: 64x32 = 512 x 16B
      int vi = i * 256 + tid;
      int r = vi >> 3, c4 = vi & 7;
      async_copy16(lds_off(&As[buf][r * 36 + c4 * 4]),
                   gaddr(&A[(size_t)(bm + r) * K + k0 + c4 * 4]));
    }
#pragma unroll
    for (int i = 0; i < 4; ++i) {     // B: 32x128 = 1024 x 16B
      int vi = i * 256 + tid;
      int r = vi >> 5, c4 = vi & 31;
      async_copy16(lds_off(&Bs[buf][r * 132 + c4 * 4]),
                   gaddr(&W[(size_t)(k0 + r) * N + bn + c4 * 4]));
    }
  };

  const int nkb = K >> 5;
  stage(0, 0);
  ASYNC_FENCE();
  __syncthreads();

  for (int kbk = 0; kbk < nkb; ++kbk) {
    const int cur = kbk & 1;
    if (kbk + 1 < nkb) stage(cur ^ 1, (kbk + 1) * 32);   // prefetch under compute

    const float* As_c = As[cur];
    const float* Bs_c = Bs[cur];
    const float* arow = &As_c[(wr * 16 + ln) * 36];

    // ---- software-pipelined fragment loads + WMMA chain ----
    v2f a_c;
    a_c.x = arow[2 * lh];
    a_c.y = arow[2 * lh + 1];
    v2f b_c[4];
#pragma unroll
    for (int t = 0; t < 4; ++t) {
      b_c[t].x = Bs_c[(lh ? 2 : 0) * 132 + wc * 64 + t * 16 + ln];
      b_c[t].y = Bs_c[(lh ? 3 : 1) * 132 + wc * 64 + t * 16 + ln];
    }
#pragma unroll
    for (int kk = 0; kk < 8; ++kk) {
      v2f a_n = a_c;
      v2f b_n[4];
      if (kk < 7) {
        const int kb = (kk + 1) * 4;
        a_n.x = arow[kb + 2 * lh];
        a_n.y = arow[kb + 2 * lh + 1];
#pragma unroll
        for (int t = 0; t < 4; ++t) {
          b_n[t].x = Bs_c[(kb + (lh ? 2 : 0)) * 132 + wc * 64 + t * 16 + ln];
          b_n[t].y = Bs_c[(kb + (lh ? 3 : 1)) * 132 + wc * 64 + t * 16 + ln];
        }
      } else {
#pragma unroll
        for (int t = 0; t < 4; ++t) b_n[t] = b_c[t];
      }
#pragma unroll
      for (int t = 0; t < 4; ++t)
        acc[t] = __builtin_amdgcn_wmma_f32_16x16x4_f32(false, a_c, false, b_c[t],
                                                       (short)0, acc[t], false, false);
      a_c = a_n;
#pragma unroll
      for (int t = 0; t < 4; ++t) b_c[t] = b_n[t];
    }

    ASYNC_FENCE();      // next buffer's DMA (in flight during compute) landed
    __syncthreads();    // all waves done reading cur before it is re-staged
  }

#pragma unroll
  for (int t = 0; t < 4; ++t) {
#pragma unroll
    for (int v = 0; v < 8; ++v) {
      int row = bm + wr * 16 + v + lh * 8;
      int col = bn + wc * 64 + t * 16 + ln;
      C[(size_t)row * N + col] = acc[t][v] + bias[col];
    }
  }
}

// ---------------------------------------------------------------------------
// Flash attention over qkv = [B, S, 3E] (q | k | v), per-head D=64.
// 128 threads = 4 waves sharing double-buffered K/V tiles (cooperative async
// staging, prefetched one tile ahead). Wave w owns q-tile qb*4+w; the block
// iterates kt uniformly to the largest needed tile, a wave skips compute for
// kt beyond its own causal frontier (<=3 idle iterations).
// Scale = 1/(sqrt(64)*2) = 1/16; masked bias = -10000 (matches reference;
// skipped tiles exact: exp(-10000-m) underflows to 0 in fp32).
// Output layout [B,S,H,D] == merged [B,S,E].
// ---------------------------------------------------------------------------
__global__ __launch_bounds__(128) void attn_kernel(const float* __restrict__ qkv,
                                                   float* __restrict__ out)
{
  __shared__ float sK[2][16 * 68];
  __shared__ float sV[2][16 * 68];
  __shared__ float sP[4][16 * 20];

  const int tid  = threadIdx.x;
  const int lane = tid & 31;
  const int ln   = lane & 15;
  const int lh   = lane >> 4;
  const int wv   = tid >> 5;

  const int bx = blockIdx.x;
  const int qb = bx & 31;               // S/64 = 32 q-blocks
  const int h  = (bx >> 5) & 15;
  const int b  = bx >> 9;
  const int qt = qb * 4 + wv;           // this wave's q tile (0..127)
  const int q0 = qt * 16;
  const int qtLast = qb * 4 + 3;        // uniform block loop bound

  float* pb = sP[wv];
  const size_t rs = 3 * (size_t)cE;     // qkv row stride in floats

  // ---- prologue: each wave parks its Q tile in one of the 4 KV buffers ----
  float* qbuf = (wv < 2) ? &sK[wv][0] : &sV[wv - 2][0];
  const float* qbase = qkv + (size_t)(b * cS + q0) * rs + h * cD;
#pragma unroll
  for (int i = 0; i < 8; ++i) {
    int vi = i * 32 + lane;
    int r = vi >> 4, c4 = vi & 15;
    async_copy16(lds_off(&qbuf[r * 68 + c4 * 4]),
                 gaddr(qbase + (size_t)r * rs + c4 * 4));
  }
  ASYNC_FENCE();

  const float qscale = 1.0f / 16.0f;
  v2f qa[16];
#pragma unroll
  for (int s = 0; s < 16; ++s) {
    qa[s].x = qbuf[ln * 68 + 4 * s + 2 * lh] * qscale;
    qa[s].y = qbuf[ln * 68 + 4 * s + 2 * lh + 1] * qscale;
  }
  LDS_FENCE();
  __syncthreads();                      // all waves extracted Q; buffers free

  // ---- cooperative KV staging (shared by all 4 waves) ----
  auto stageKV = [&](int buf, int kt_) {
    const float* kt0 = qkv + (size_t)(b * cS + kt_ * 16) * rs + cE + h * cD;
    const float* vt0 = kt0 + cE;
#pragma unroll
    for (int i = 0; i < 2; ++i) {       // 16x64 = 256 x 16B each, 128 threads
      int vi = i * 128 + tid;
      int r = vi >> 4, c4 = vi & 15;
      async_copy16(lds_off(&sK[buf][r * 68 + c4 * 4]),
                   gaddr(kt0 + (size_t)r * rs + c4 * 4));
      async_copy16(lds_off(&sV[buf][r * 68 + c4 * 4]),
                   gaddr(vt0 + (size_t)r * rs + c4 * 4));
    }
  };
  stageKV(0, 0);
  ASYNC_FENCE();
  __syncthreads();

  v8f o[4] = {};
  float mrun[8], lrun[8];
#pragma unroll
  for (int v = 0; v < 8; ++v) { mrun[v] = -3.0e38f; lrun[v] = 0.0f; }

  for (int kt = 0; kt <= qtLast; ++kt) {
    const int cur = kt & 1;
    if (kt + 1 <= qtLast) stageKV(cur ^ 1, kt + 1);   // prefetch under compute

    if (kt <= qt) {
      const float* kb = sK[cur];
      const float* vb = sV[cur];

      // ---- S = Qs @ K^T (pipelined B-fragment loads) ----
      v8f sacc = {};
      v2f bf_c;
      bf_c.x = kb[ln * 68 + (lh ? 2 : 0)];
      bf_c.y = kb[ln * 68 + (lh ? 3 : 1)];
#pragma unroll
      for (int s = 0; s < 16; ++s) {
        v2f bf_n = bf_c;
        if (s < 15) {
          bf_n.x = kb[ln * 68 + 4 * (s + 1) + (lh ? 2 : 0)];
          bf_n.y = kb[ln * 68 + 4 * (s + 1) + (lh ? 3 : 1)];
        }
        sacc = __builtin_amdgcn_wmma_f32_16x16x4_f32(false, qa[s], false, bf_c,
                                                     (short)0, sacc, false, false);
        bf_c = bf_n;
      }

      // ---- causal mask on diagonal tile (kv > q -> -10000) ----
      if (kt == qt) {
#pragma unroll
        for (int v = 0; v < 8; ++v) {
          int qrow = v + 8 * lh;
          if (ln > qrow) sacc[v] = -10000.0f;
        }
      }

      // ---- online softmax (row = (vgpr, lane-half); 16-lane reductions) ----
#pragma unroll
      for (int v = 0; v < 8; ++v) {
        float x  = sacc[v];
        float tm = x;
        tm = fmaxf(tm, __shfl_xor(tm, 1, 32));
        tm = fmaxf(tm, __shfl_xor(tm, 2, 32));
        tm = fmaxf(tm, __shfl_xor(tm, 4, 32));
        tm = fmaxf(tm, __shfl_xor(tm, 8, 32));
        float mnew = fmaxf(mrun[v], tm);
        float p    = __expf(x - mnew);
        float corr = __expf(mrun[v] - mnew);
        float ps = p;
        ps += __shfl_xor(ps, 1, 32);
        ps += __shfl_xor(ps, 2, 32);
        ps += __shfl_xor(ps, 4, 32);
        ps += __shfl_xor(ps, 8, 32);
        lrun[v] = lrun[v] * corr + ps;
        mrun[v] = mnew;
        o[0][v] *= corr; o[1][v] *= corr; o[2][v] *= corr; o[3][v] *= corr;
        pb[(v + 8 * lh) * 20 + ln] = p;  // C-layout -> LDS for A-fragments
      }
      LDS_FENCE();

      // ---- O += P @ V (pipelined fragment loads) ----
      const int r0b = (lh ? 2 : 0), r1b = (lh ? 3 : 1);
      v2f a_c;
      a_c.x = pb[ln * 20 + 2 * lh];
      a_c.y = pb[ln * 20 + 2 * lh + 1];
      v2f vb_c[4];
#pragma unroll
      for (int t = 0; t < 4; ++t) {
        vb_c[t].x = vb[r0b * 68 + t * 16 + ln];
        vb_c[t].y = vb[r1b * 68 + t * 16 + ln];
      }
#pragma unroll
      for (int s = 0; s < 4; ++s) {
        v2f a_n = a_c;
        v2f vb_n[4];
        if (s < 3) {
          const int kb4 = (s + 1) * 4;
          a_n.x = pb[ln * 20 + kb4 + 2 * lh];
          a_n.y = pb[ln * 20 + kb4 + 2 * lh + 1];
#pragma unroll
          for (int t = 0; t < 4; ++t) {
            vb_n[t].x = vb[(kb4 + r0b) * 68 + t * 16 + ln];
            vb_n[t].y = vb[(kb4 + r1b) * 68 + t * 16 + ln];
          }
        } else {
#pragma unroll
          for (int t = 0; t < 4; ++t) vb_n[t] = vb_c[t];
        }
#pragma unroll
        for (int t = 0; t < 4; ++t)
          o[t] = __builtin_amdgcn_wmma_f32_16x16x4_f32(false, a_c, false, vb_c[t],
                                                       (short)0, o[t], false, false);
        a_c = a_n;
#pragma unroll
        for (int t = 0; t < 4; ++t) vb_c[t] = vb_n[t];
      }
    }

    LDS_FENCE();        // this wave's ds reads of cur done
    ASYNC_FENCE();      // this wave's share of next buffer's DMA landed
    __syncthreads();    // whole block in sync before cur is re-staged
  }

  // ---- finalize: divide by row sums, write merged-head layout ----
#pragma unroll
  for (int v = 0; v < 8; ++v) {
    float inv = 1.0f / lrun[v];
    int q = q0 + v + 8 * lh;
    float* orow = out + (size_t)(b * cS + q) * cE + h * cD;
    orow[ln]      = o[0][v] * inv;
    orow[16 + ln] = o[1][v] * inv;
    orow[32 + ln] = o[2][v] * inv;
    orow[48 + ln] = o[3][v] * inv;
  }
}

// ---------------------------------------------------------------------------
extern "C" void kernel_launch(void* const* d_in, const int* in_sizes, int n_in,
                              void* d_out, int out_size, void* d_ws, size_t ws_size,
                              hipStream_t stream)
{
  (void)in_sizes; (void)n_in; (void)out_size; (void)ws_size;
  const float* hidden = (const float*)d_in[0];   // [B,S,E]
  const float* w_attn = (const float*)d_in[1];   // [E,3E]
  const float* b_attn = (const float*)d_in[2];   // [3E]
  const float* w_proj = (const float*)d_in[3];   // [E,E]
  const float* b_proj = (const float*)d_in[4];   // [E]
  float* C = (float*)d_out;                      // [B,S,E]

  const int M = cB * cS;                         // 4096
  float* qkv      = (float*)d_ws;                // M * 3E floats (50.3 MB)
  float* attn_out = qkv + (size_t)M * 3 * cE;    // M * E floats  (16.8 MB)

  // 1) QKV GEMM: [4096,1024] @ [1024,3072] + b
  dim3 g1(3 * cE / 128, M / 64);
  gemm_bias_kernel<<<g1, 256, 0, stream>>>(hidden, w_attn, b_attn, qkv, M, 3 * cE, cE);

  // 2) Flash attention: B*H*(S/64) blocks of 4 waves
  attn_kernel<<<cB * cH * (cS / 64), 128, 0, stream>>>(qkv, attn_out);

  // 3) Proj GEMM: [4096,1024] @ [1024,1024] + b -> d_out
  dim3 g2(cE / 128, M / 64);
  gemm_bias_kernel<<<g2, 256, 0, stream>>>(attn_out, w_proj, b_proj, C, M, cE, cE);
}